// KeypointFlowLoss_28767690949028
// MI455X (gfx1250) — compile-verified
//
#include <hip/hip_runtime.h>

#define NL 4
#define NB 64
#define NK 17
#define NH 256
#define NW 256
#define NTHREADS 1024

typedef int v4i __attribute__((vector_size(4 * sizeof(int))));
typedef __attribute__((address_space(1))) v4i* gv4i_p;   // global
typedef __attribute__((address_space(3))) v4i* lv4i_p;   // LDS

__global__ __launch_bounds__(NTHREADS)
void KeypointFlowLoss_28767690949028_kernel(const float* __restrict__ pred,
                                            const int* __restrict__ kps,
                                            float* __restrict__ out)
{
    // kps staged in LDS: (B, 2, K, 2) int32 = 17408 bytes
    __shared__ __align__(16) int skps[NB * 2 * NK * 2];
    __shared__ float red_sse[NTHREADS / 32];
    __shared__ float red_cnt[NTHREADS / 32];

    const int tid = threadIdx.x;
    constexpr int KPS_INTS = NB * 2 * NK * 2;   // 4352 ints
    constexpr int NCHUNK   = KPS_INTS / 4;      // 1088 x 16B chunks

#if __has_builtin(__builtin_amdgcn_global_load_async_to_lds_b128)
    // CDNA5 async global->LDS copy, tracked by ASYNCcnt (no VGPR data path).
    for (int c = tid; c < NCHUNK; c += NTHREADS) {
        __builtin_amdgcn_global_load_async_to_lds_b128(
            (gv4i_p)(kps + c * 4),
            (lv4i_p)(&skps[c * 4]),
            /*offset=*/0, /*cpol=*/0);
    }
#if __has_builtin(__builtin_amdgcn_s_wait_asynccnt)
    __builtin_amdgcn_s_wait_asynccnt(0);
#else
    asm volatile("s_wait_asynccnt 0" ::: "memory");
#endif
#else
    // Fallback: plain staged copy.
    for (int c = tid; c < KPS_INTS; c += NTHREADS) skps[c] = kps[c];
#endif
    __syncthreads();

    // Each (l,b,k) tuple: gather 2 pred values, accumulate weighted SSE.
    float wsse = 0.0f;
    float fcnt = 0.0f;
    constexpr int TOTAL = NL * NB * NK;  // 4352
    for (int idx = tid; idx < TOTAL; idx += NTHREADS) {
        const int l = idx / (NB * NK);
        const int r = idx - l * (NB * NK);
        const int b = r / NK;
        const int k = r - b * NK;

        const int o0 = (b * 2 + 0) * (NK * 2) + k * 2;
        const int o1 = (b * 2 + 1) * (NK * 2) + k * 2;
        const int x0 = skps[o0 + 0];
        const int y0 = skps[o0 + 1];
        const int x1 = skps[o1 + 0];
        const int y1 = skps[o1 + 1];
        const int idx_dx = x1 - x0;
        const int idx_dy = y1 - y0;

        if ((idx_dx | idx_dy) != 0) {   // mask: sqrt(dx^2+dy^2) > 0
            const float dx = (float)idx_dx;
            const float dy = (float)idx_dy;
            const size_t base =
                ((((size_t)l * NB + (size_t)b) * 2) * NH + (size_t)y0) * NW + (size_t)x0;
            const float p0 = pred[base];                      // channel 0
            const float p1 = pred[base + (size_t)NH * NW];    // channel 1
            const float e0 = p0 - dx;
            const float e1 = p1 - dy;
            // gamma^(L-1-l), gamma = 0.8
            const float w = (l == 0) ? 0.512f
                          : (l == 1) ? 0.64f
                          : (l == 2) ? 0.8f
                                     : 1.0f;
            wsse = fmaf(w, e0 * e0 + e1 * e1, wsse);
            if (l == 0) fcnt += 1.0f;   // count each valid keypoint exactly once
        }
    }

    // Deterministic reduction: wave32 butterfly, then cross-wave via LDS.
    for (int m = 16; m > 0; m >>= 1) {
        wsse += __shfl_xor(wsse, m, 32);
        fcnt += __shfl_xor(fcnt, m, 32);
    }
    const int wave = tid >> 5;
    const int lane = tid & 31;
    if (lane == 0) {
        red_sse[wave] = wsse;
        red_cnt[wave] = fcnt;
    }
    __syncthreads();
    if (wave == 0) {
        float s = red_sse[lane];
        float c = red_cnt[lane];
        for (int m = 16; m > 0; m >>= 1) {
            s += __shfl_xor(s, m, 32);
            c += __shfl_xor(c, m, 32);
        }
        if (lane == 0) {
            const float denom = fmaxf(c * 2.0f, 1.0f);   // sum(maskf) * C, clamped
            out[0] = s / denom;                           // LOSS_WEIGHT = 1
        }
    }
}

extern "C" void kernel_launch(void* const* d_in, const int* in_sizes, int n_in,
                              void* d_out, int out_size, void* d_ws, size_t ws_size,
                              hipStream_t stream) {
    const float* pred = (const float*)d_in[0];   // (4, 64, 2, 256, 256) fp32
    const int*   kps  = (const int*)d_in[1];     // (64, 2, 17, 2) int32
    float*       out  = (float*)d_out;           // scalar loss

    KeypointFlowLoss_28767690949028_kernel<<<dim3(1), dim3(NTHREADS), 0, stream>>>(
        pred, kps, out);
}